// PredictiveCodingNet_64063732187565
// MI455X (gfx1250) — compile-verified
//
#include <hip/hip_runtime.h>
#include <cstddef>

// CDNA5 / gfx1250: wave32, fp32 WMMA path (V_WMMA_F32_16X16X4_F32).
// Each wave computes a 32x32 output tile with 4 WMMA accumulators; both GEMM
// flavors use identical contiguous-b64 fragment loads (weights for the
// "update" GEMM are pre-transposed once per launch via an LDS-tiled kernel).
typedef float v2f __attribute__((ext_vector_type(2)));
typedef float v8f __attribute__((ext_vector_type(8)));

#define PC_LR 0.05f

__device__ __forceinline__ v8f wmma_f32x4(v2f a, v2f b, v8f c) {
  // 8-arg form: (neg_a, A, neg_b, B, c_mod, C, reuse_a, reuse_b)
  return __builtin_amdgcn_wmma_f32_16x16x4_f32(false, a, false, b, (short)0, c,
                                               false, false);
}

// Shared 32x32-tile GEMM core: per K-step of 4, load 2 A + 2 B fragments
// (contiguous b64 each) and issue 4 v_wmma on resident accumulators.
// a0/a1: two 16-row A fragment pointers (row stride K, pre-offset by khalf)
// b0/b1: two 16-col B fragment pointers (row stride K, pre-offset by khalf)
__device__ __forceinline__ void gemm_core_32x32(const float* __restrict__ a0,
                                                const float* __restrict__ a1,
                                                const float* __restrict__ b0,
                                                const float* __restrict__ b1,
                                                int K, v8f acc[2][2]) {
  for (int k = 0; k < K; k += 16) {
#pragma unroll
    for (int kk = 0; kk < 16; kk += 4) {
      const v2f A0 = *(const v2f*)(a0 + k + kk);   // global_load_b64
      const v2f A1 = *(const v2f*)(a1 + k + kk);
      const v2f B0 = *(const v2f*)(b0 + k + kk);
      const v2f B1 = *(const v2f*)(b1 + k + kk);
      acc[0][0] = wmma_f32x4(A0, B0, acc[0][0]);
      acc[0][1] = wmma_f32x4(A0, B1, acc[0][1]);
      acc[1][0] = wmma_f32x4(A1, B0, acc[1][0]);
      acc[1][1] = wmma_f32x4(A1, B1, acc[1][1]);
    }
  }
}

// ---------------------------------------------------------------------------
// Weight transpose (runs once per launch, outside the 50-step loop).
//   in : (R x C) row-major  ->  out : (C x R) row-major, out[c][r] = in[r][c]
// LDS 32x33 tile (pad kills bank conflicts); R, C multiples of 32.
// ---------------------------------------------------------------------------
__global__ void __launch_bounds__(256)
pc_transpose_kernel(const float* __restrict__ in, float* __restrict__ out,
                    int R, int C)
{
  __shared__ float tile[32][33];
  const int tc = blockIdx.x << 5;          // column base in `in`
  const int tr = blockIdx.y << 5;          // row base in `in`
  const int lx = threadIdx.x & 31;
  const int ly = threadIdx.x >> 5;         // 0..7
#pragma unroll
  for (int yy = 0; yy < 32; yy += 8)
    tile[ly + yy][lx] = in[(size_t)(tr + ly + yy) * C + (tc + lx)];
  __syncthreads();
#pragma unroll
  for (int yy = 0; yy < 32; yy += 8)
    out[(size_t)(tc + ly + yy) * R + (tr + lx)] = tile[lx][ly + yy];
}

// ---------------------------------------------------------------------------
// eps = cur - tanh(act @ W^T + bias)          (predict + error, fused)
//   act : (Btot x K) row-major   (A, rows = batch)
//   W   : (M x K)    row-major   (B = W^T: W rows = output cols)
// If upd_scale != 0, also in-place cur -= upd_scale*eps (fused r0 step).
// ---------------------------------------------------------------------------
__global__ void __launch_bounds__(128)
pc_eps_kernel(const float* __restrict__ act, const float* __restrict__ W,
              const float* __restrict__ bias, float* __restrict__ cur,
              float* __restrict__ eps, int Btot, int K, int M, float upd_scale)
{
  const int wave   = threadIdx.x >> 5;
  const int lane   = threadIdx.x & 31;
  const int tilesM = M >> 5;
  const int tile   = blockIdx.x * 4 + wave;
  if (tile >= (Btot >> 5) * tilesM) return;   // wave-uniform: EXEC all-1s
  const int tb = tile / tilesM;
  const int tm = tile % tilesM;

  const int n     = lane & 15;                // index within 16-wide sub-tile
  const int khalf = (lane >> 4) << 1;         // lanes 0-15: K 0,1; 16-31: K 2,3

  const float* a0 = act + (size_t)((tb << 5) + n) * K + khalf;
  const float* a1 = a0 + (size_t)16 * K;
  const float* b0 = W + (size_t)((tm << 5) + n) * K + khalf;
  const float* b1 = b0 + (size_t)16 * K;

  v8f acc[2][2] = {};
  gemm_core_32x32(a0, a1, b0, b1, K, acc);

  // D layout per 16x16 sub-tile: lane, vgpr v -> row v+8*(lane/16), col lane%16
#pragma unroll
  for (int i = 0; i < 2; ++i) {
    const int rbase = (tb << 5) + (i << 4) + ((lane >> 4) << 3);
#pragma unroll
    for (int j = 0; j < 2; ++j) {
      const int colm = (tm << 5) + (j << 4) + n;
      const float bv = bias[colm];
#pragma unroll
      for (int v = 0; v < 8; ++v) {
        const size_t idx = (size_t)(rbase + v) * M + colm;
        const float p = tanhf(acc[i][j][v] + bv);
        const float c = cur[idx];
        const float e = c - p;
        eps[idx] = e;
        if (upd_scale != 0.0f) cur[idx] = c - upd_scale * e; // fused r0 update
      }
    }
  }
}

// ---------------------------------------------------------------------------
// r += LR * ( -eps_self + eps_prev @ W )      (propagate + state update, fused)
//   eps_prev : (Btot x K) row-major           (A matrix)
//   WT       : (M x K)    row-major           (pre-transposed W: WT[m][k]=W[k][m])
//   eps_self : (Btot x M) or nullptr (r3 has no -eps term)
// Same contiguous-b64 fragment loads as the predict kernel.
// ---------------------------------------------------------------------------
__global__ void __launch_bounds__(128)
pc_update_kernel(const float* __restrict__ eps_prev, const float* __restrict__ WT,
                 const float* __restrict__ eps_self, float* __restrict__ r,
                 int Btot, int K, int M)
{
  const int wave   = threadIdx.x >> 5;
  const int lane   = threadIdx.x & 31;
  const int tilesM = M >> 5;
  const int tile   = blockIdx.x * 4 + wave;
  if (tile >= (Btot >> 5) * tilesM) return;
  const int tb = tile / tilesM;
  const int tm = tile % tilesM;

  const int n     = lane & 15;
  const int khalf = (lane >> 4) << 1;

  const float* a0 = eps_prev + (size_t)((tb << 5) + n) * K + khalf;
  const float* a1 = a0 + (size_t)16 * K;
  const float* b0 = WT + (size_t)((tm << 5) + n) * K + khalf;  // WT rows = cols
  const float* b1 = b0 + (size_t)16 * K;

  v8f acc[2][2] = {};
  gemm_core_32x32(a0, a1, b0, b1, K, acc);

#pragma unroll
  for (int i = 0; i < 2; ++i) {
    const int rbase = (tb << 5) + (i << 4) + ((lane >> 4) << 3);
#pragma unroll
    for (int j = 0; j < 2; ++j) {
      const int col = (tm << 5) + (j << 4) + n;
#pragma unroll
      for (int v = 0; v < 8; ++v) {
        const size_t idx = (size_t)(rbase + v) * M + col;
        float upd = acc[i][j][v];
        if (eps_self) upd -= eps_self[idx];
        r[idx] += PC_LR * upd;
      }
    }
  }
}

// ---------------------------------------------------------------------------

extern "C" void kernel_launch(void* const* d_in, const int* in_sizes, int n_in,
                              void* d_out, int out_size, void* d_ws, size_t ws_size,
                              hipStream_t stream) {
  (void)in_sizes; (void)n_in; (void)out_size; (void)ws_size;
  const int B = 8192, d1 = 1024, d2 = 512, d3 = 256, d4 = 128;
  const int T_INFER = 50;

  const float* layer1 = (const float*)d_in[0];
  const float* layer2 = (const float*)d_in[1];
  const float* layer3 = (const float*)d_in[2];
  const float* layer4 = (const float*)d_in[3];
  const float* W1     = (const float*)d_in[4];   // (d1, d2)
  const float* W2     = (const float*)d_in[5];   // (d2, d3)
  const float* W3     = (const float*)d_in[6];   // (d3, d4)
  const float* b1     = (const float*)d_in[7];
  const float* b2     = (const float*)d_in[8];
  const float* b3     = (const float*)d_in[9];

  // State lives directly in d_out (outputs are r0..r3 concatenated flat).
  float* r0 = (float*)d_out;
  float* r1 = r0 + (size_t)B * d1;
  float* r2 = r1 + (size_t)B * d2;
  float* r3 = r2 + (size_t)B * d3;

  // Workspace: eps scratch (~59 MB) + transposed weights (~2.6 MB); all
  // L2-resident on MI455X (192 MB global L2).
  float* eps0 = (float*)d_ws;
  float* eps1 = eps0 + (size_t)B * d1;
  float* eps2 = eps1 + (size_t)B * d2;
  float* WT1  = eps2 + (size_t)B * d3;           // (d2 x d1)
  float* WT2  = WT1 + (size_t)d2 * d1;           // (d3 x d2)
  float* WT3  = WT2 + (size_t)d3 * d2;           // (d4 x d3)

  // Seed state from inputs (d2d async copies are graph-capture-safe).
  hipMemcpyAsync(r0, layer1, (size_t)B * d1 * sizeof(float), hipMemcpyDeviceToDevice, stream);
  hipMemcpyAsync(r1, layer2, (size_t)B * d2 * sizeof(float), hipMemcpyDeviceToDevice, stream);
  hipMemcpyAsync(r2, layer3, (size_t)B * d3 * sizeof(float), hipMemcpyDeviceToDevice, stream);
  hipMemcpyAsync(r3, layer4, (size_t)B * d4 * sizeof(float), hipMemcpyDeviceToDevice, stream);

  // One-time weight transposes (tiny; outside the step loop).
  pc_transpose_kernel<<<dim3(d2 / 32, d1 / 32), 256, 0, stream>>>(W1, WT1, d1, d2);
  pc_transpose_kernel<<<dim3(d3 / 32, d2 / 32), 256, 0, stream>>>(W2, WT2, d2, d3);
  pc_transpose_kernel<<<dim3(d4 / 32, d3 / 32), 256, 0, stream>>>(W3, WT3, d3, d4);

  auto nblocks = [](int Btot, int M) {
    return (((Btot >> 5) * (M >> 5)) + 3) / 4;   // 4 waves (32x32 tiles)/block
  };
  const dim3 blk(128);

  for (int t = 0; t < T_INFER; ++t) {
    // Phase 1: errors against OLD state (stream order guarantees correctness).
    // eps0 = r0 - tanh(r1 @ W1^T + b1); fused r0 -= 0.5*LR*eps0
    pc_eps_kernel<<<nblocks(B, d1), blk, 0, stream>>>(r1, W1, b1, r0, eps0,
                                                      B, d2, d1, 0.5f * PC_LR);
    // eps1 = r1 - tanh(r2 @ W2^T + b2)
    pc_eps_kernel<<<nblocks(B, d2), blk, 0, stream>>>(r2, W2, b2, r1, eps1,
                                                      B, d3, d2, 0.0f);
    // eps2 = r2 - tanh(r3 @ W3^T + b3)
    pc_eps_kernel<<<nblocks(B, d3), blk, 0, stream>>>(r3, W3, b3, r2, eps2,
                                                      B, d4, d3, 0.0f);

    // Phase 2: state updates (consume eps written above).
    // r1 += LR*(-eps1 + eps0 @ W1)     (WT1 = W1^T)
    pc_update_kernel<<<nblocks(B, d2), blk, 0, stream>>>(eps0, WT1, eps1, r1,
                                                         B, d1, d2);
    // r2 += LR*(-eps2 + eps1 @ W2)     (WT2 = W2^T)
    pc_update_kernel<<<nblocks(B, d3), blk, 0, stream>>>(eps1, WT2, eps2, r2,
                                                         B, d2, d3);
    // r3 += LR*(eps2 @ W3)             (WT3 = W3^T)
    pc_update_kernel<<<nblocks(B, d4), blk, 0, stream>>>(eps2, WT3, nullptr, r3,
                                                         B, d3, d4);
  }
}